// RelationLoss_87754771792036
// MI455X (gfx1250) — compile-verified
//
#include <hip/hip_runtime.h>
#include <hip/hip_bf16.h>
#include <stdint.h>

// ---------------------------------------------------------------------------
// Problem constants (match reference): B=8, N=2048, D=128, k = 0.1*N*N
// ---------------------------------------------------------------------------
#define BATCH 8
#define NDIM  2048
#define DDIM  128
#define NN    (NDIM * NDIM)   // 4,194,304 elements per sim matrix

typedef __attribute__((ext_vector_type(2))) float v2f;
typedef __attribute__((ext_vector_type(8))) float v8f;

// Workspace layout (bytes):
//   [0)                : sims: 2 * NN floats (which=0 input sim, which=1 target sim)
//   [SIM_BYTES)        : hist: 2 * 2048 uint32
//   [+16384)           : state: 2 * 4 uint32  {prefix, rank, thr_bits, pad}
//   [+64)              : partials: BATCH * MSE_BLOCKS floats
#define SIM_BYTES   ((size_t)2 * NN * 4)
#define HIST_OFF    SIM_BYTES
#define STATE_OFF   (HIST_OFF + (size_t)2 * 2048 * 4)
#define PART_OFF    (STATE_OFF + 64)
#define MSE_BLOCKS  512

// Order-preserving key for f32: ascending key <=> ascending float
__device__ __forceinline__ uint32_t fkey(float f) {
  uint32_t u = __float_as_uint(f);
  return (u & 0x80000000u) ? ~u : (u | 0x80000000u);
}

// Async direct global->LDS copy (ASYNCcnt path, gfx1250).
// lds_off is the absolute LDS byte offset (dynamic-LDS base == 0 here).
__device__ __forceinline__ void async_load_b128(uint32_t lds_off, const float* g) {
  asm volatile("global_load_async_to_lds_b128 %0, %1, off"
               :: "v"(lds_off), "v"((uint64_t)(uintptr_t)g)
               : "memory");
}
__device__ __forceinline__ void wait_async0() {
  asm volatile("s_wait_asynccnt 0" ::: "memory");
}

// ---------------------------------------------------------------------------
// Kernel 1: sim = X @ X^T for one batch, which = blockIdx.z (0=input,1=target)
// 256 threads = 8 waves. Block computes a 128x128 tile (upper triangle only;
// the mirror tile is stored transposed — X X^T is bitwise symmetric since the
// k-order of the reduction is identical for (i,j) and (j,i)).
// K is staged through double-buffered LDS chunks filled with
// GLOBAL_LOAD_ASYNC_TO_LDS_B128 so the fill of chunk kc+1 overlaps the
// v_wmma_f32_16x16x4_f32 work on chunk kc.
// ---------------------------------------------------------------------------
#define KCHUNK 16
#define LDSP   20          // padded row stride (floats): conflict-free ds_load_b64
#define ABUF_FLOATS (2 * 128 * LDSP)           // two A buffers
#define GEMM_LDS_BYTES ((size_t)2 * ABUF_FLOATS * 4)  // A + B regions = 40960 B

__global__ __launch_bounds__(256) void gemm_sim(
    const float* __restrict__ inp,   // this batch's input  rows [N x D]
    const float* __restrict__ tgt,   // this batch's target rows [N x D]
    float* __restrict__ sims,        // ws: 2 * NN floats
    uint32_t* __restrict__ state,    // ws: select state (2 x 4 u32)
    const int* __restrict__ kptr)    // device scalar k
{
  const int ti = blockIdx.y;        // row tile
  const int tj = blockIdx.x;        // col tile
  const int tid = threadIdx.x;

  // Reset radix-select state for this (batch, which) once per batch
  if (ti == 0 && tj == 0 && tid == 0) {
    const int which0 = blockIdx.z;
    state[which0 * 4 + 0] = 0u;                  // prefix
    state[which0 * 4 + 1] = (uint32_t)(*kptr);   // remaining rank (k-th largest)
  }
  if (tj < ti) return;              // symmetric: compute upper triangle only

  const int which = blockIdx.z;
  const float* X  = which ? tgt : inp;
  float* sim      = sims + (size_t)which * NN;

  extern __shared__ float smem[];   // [A: 2*128*LDSP][B: 2*128*LDSP]
  float* Abuf = smem;
  float* Bbuf = smem + ABUF_FLOATS;

  const int i0 = ti * 128;          // output rows
  const int j0 = tj * 128;          // output cols

  const int wave = tid >> 5;        // 0..7
  const int lane = tid & 31;
  const int lm   = lane & 15;       // M (or N) index within fragment
  const int half = lane >> 4;       // K pair selector
  const int m_sub = wave * 16;

  // Per-thread async fill: 2 float4 for A + 2 float4 for B per 128x16 chunk
  const int l_row  = tid >> 2;            // rows 0..63 (t=0), 64..127 (t=1)
  const int l_col  = (tid & 3) * 4;       // 0,4,8,12

  auto issue_chunk = [&](int kc, int buf) {
    const int kbase = kc * KCHUNK;
#pragma unroll
    for (int t = 0; t < 2; ++t) {
      const int row = l_row + t * 64;
      const uint32_t lofs = (uint32_t)(((buf * 128 + row) * LDSP + l_col) * 4);
      async_load_b128(lofs,
                      X + (size_t)(i0 + row) * DDIM + kbase + l_col);
      async_load_b128(lofs + (uint32_t)(ABUF_FLOATS * 4),
                      X + (size_t)(j0 + row) * DDIM + kbase + l_col);
    }
  };

  v8f acc[8];
#pragma unroll
  for (int n = 0; n < 8; ++n) acc[n] = (v8f)0.0f;

  issue_chunk(0, 0);
  wait_async0();
  __syncthreads();

  for (int kc = 0; kc < DDIM / KCHUNK; ++kc) {
    const int buf = kc & 1;
    if (kc + 1 < DDIM / KCHUNK) issue_chunk(kc + 1, buf ^ 1);   // prefetch

    const int arow = (buf * 128 + m_sub + lm) * LDSP;
#pragma unroll
    for (int ks = 0; ks < KCHUNK / 4; ++ks) {
      const int kk = ks * 4 + 2 * half;
      // A fragment: lane<16 holds A[M=lm][K=kk..kk+1], lane>=16 K=kk+2..kk+3
      v2f a = *(const v2f*)&Abuf[arow + kk];
#pragma unroll
      for (int n = 0; n < 8; ++n) {
        // B fragment: B[k][n] = X[j0 + n][k] (B rows striped across lanes)
        v2f b = *(const v2f*)&Bbuf[(buf * 128 + n * 16 + lm) * LDSP + kk];
        acc[n] = __builtin_amdgcn_wmma_f32_16x16x4_f32(
            false, a, false, b, (short)0, acc[n], false, false);
      }
    }

    if (kc + 1 < DDIM / KCHUNK) wait_async0();  // next chunk landed
    __syncthreads();                            // all waves done with this buf
  }

  // Write back: C VGPR j, lane l -> row = j + 8*half, col = lm
#pragma unroll
  for (int n = 0; n < 8; ++n) {
    const int col = j0 + n * 16 + lm;
#pragma unroll
    for (int j = 0; j < 8; ++j) {
      const int row = i0 + m_sub + j + 8 * half;
      sim[(size_t)row * NDIM + col] = acc[n][j];
    }
  }
  if (tj > ti) {  // mirror tile (exactly equal by symmetry); L2 absorbs scatter
#pragma unroll
    for (int n = 0; n < 8; ++n) {
      const int col = j0 + n * 16 + lm;
#pragma unroll
      for (int j = 0; j < 8; ++j) {
        const int row = i0 + m_sub + j + 8 * half;
        sim[(size_t)col * NDIM + row] = acc[n][j];
      }
    }
  }
}

// ---------------------------------------------------------------------------
// Kernel 2: radix-select histogram pass. pass 0: key[31:21] (2048 bins),
// pass 1: key[20:10] filtered by 11-bit prefix, pass 2: key[9:0] (1024 bins)
// filtered by 22-bit prefix. which = blockIdx.y.
// ---------------------------------------------------------------------------
__global__ __launch_bounds__(256) void hist_pass(
    const float* __restrict__ sims,
    uint32_t* __restrict__ hist,
    const uint32_t* __restrict__ state,
    int pass)
{
  const int which = blockIdx.y;
  const float* sim = sims + (size_t)which * NN;
  const uint32_t prefix = state[which * 4 + 0];

  __shared__ uint32_t lh[2048];
  for (int i = threadIdx.x; i < 2048; i += blockDim.x) lh[i] = 0u;
  __syncthreads();

  const int stride = gridDim.x * blockDim.x;
  for (int i = blockIdx.x * blockDim.x + threadIdx.x; i < NN; i += stride) {
    uint32_t key = fkey(sim[i]);
    if (pass == 0) {
      atomicAdd(&lh[key >> 21], 1u);
    } else if (pass == 1) {
      if ((key >> 21) == prefix) atomicAdd(&lh[(key >> 10) & 2047u], 1u);
    } else {
      if ((key >> 10) == prefix) atomicAdd(&lh[key & 1023u], 1u);
    }
  }
  __syncthreads();
  for (int i = threadIdx.x; i < 2048; i += blockDim.x) {
    uint32_t c = lh[i];
    if (c) atomicAdd(&hist[which * 2048 + i], c);
  }
}

// ---------------------------------------------------------------------------
// Kernel 3: pick the bin holding the rank-th largest, update prefix/rank,
// and zero the histogram for the next pass/batch. One block per `which`.
// ---------------------------------------------------------------------------
__global__ __launch_bounds__(64) void select_bin(
    uint32_t* __restrict__ hist,
    uint32_t* __restrict__ state,
    int pass)
{
  const int which = blockIdx.x;
  if (threadIdx.x == 0) {
    uint32_t prefix = state[which * 4 + 0];
    uint32_t rank   = state[which * 4 + 1];   // 1-based rank from the top
    const int nb = (pass == 2) ? 1024 : 2048;
    uint32_t cum = 0;
    int sel = 0;
    for (int bin = nb - 1; bin >= 0; --bin) {
      uint32_t c = hist[which * 2048 + bin];
      if (cum + c >= rank) { sel = bin; rank -= cum; break; }
      cum += c;
    }
    if (pass < 2) {
      state[which * 4 + 0] = (prefix << 11) | (uint32_t)sel;
      state[which * 4 + 1] = rank;
    } else {
      uint32_t key = (prefix << 10) | (uint32_t)sel;  // exact key of threshold
      uint32_t u = (key & 0x80000000u) ? (key ^ 0x80000000u) : ~key;
      state[which * 4 + 2] = u;                        // thr bits (float)
    }
  }
  __syncthreads();
  for (int i = threadIdx.x; i < 2048; i += blockDim.x)
    hist[which * 2048 + i] = 0u;
}

// ---------------------------------------------------------------------------
// Kernel 4: masked MSE partial sums for one batch (deterministic: no float
// atomics; one partial per block).
// ---------------------------------------------------------------------------
__global__ __launch_bounds__(256) void mse_pass(
    const float* __restrict__ sims,
    const uint32_t* __restrict__ state,
    float* __restrict__ partial)   // MSE_BLOCKS floats for this batch
{
  const float thr0 = __uint_as_float(state[0 * 4 + 2]);
  const float thr1 = __uint_as_float(state[1 * 4 + 2]);

  float s = 0.0f;
  const int stride = gridDim.x * blockDim.x;
  for (int i = blockIdx.x * blockDim.x + threadIdx.x; i < NN; i += stride) {
    float a = sims[i];
    float b = sims[(size_t)NN + i];
    a = (a < thr0) ? 0.0f : a;
    b = (b < thr1) ? 0.0f : b;
    float d = a - b;
    s += d * d;
  }

  __shared__ float red[256];
  red[threadIdx.x] = s;
  __syncthreads();
#pragma unroll
  for (int off = 128; off > 0; off >>= 1) {
    if (threadIdx.x < off) red[threadIdx.x] += red[threadIdx.x + off];
    __syncthreads();
  }
  if (threadIdx.x == 0) partial[blockIdx.x] = red[0];
}

// ---------------------------------------------------------------------------
// Kernel 5: final reduction -> scalar mean
// ---------------------------------------------------------------------------
__global__ void finalize(const float* __restrict__ partial, float* __restrict__ out)
{
  if (threadIdx.x == 0 && blockIdx.x == 0) {
    double acc = 0.0;
    for (int i = 0; i < BATCH * MSE_BLOCKS; ++i) acc += (double)partial[i];
    out[0] = (float)(acc / ((double)BATCH * (double)NN));
  }
}

// Zero both histograms once at the start of each launch (ws is poisoned).
__global__ void zero_hist(uint32_t* __restrict__ hist)
{
  for (int i = threadIdx.x; i < 2 * 2048; i += blockDim.x) hist[i] = 0u;
}

// ---------------------------------------------------------------------------
extern "C" void kernel_launch(void* const* d_in, const int* in_sizes, int n_in,
                              void* d_out, int out_size, void* d_ws, size_t ws_size,
                              hipStream_t stream)
{
  const float* inp = (const float*)d_in[0];   // [B, N, D] f32
  const float* tgt = (const float*)d_in[1];   // [B, N, D] f32
  // d_in[2] = indices: only its length matters in the reference -> unused
  const int* kptr  = (const int*)d_in[3];     // scalar k (419430)
  float* out       = (float*)d_out;

  float*    sims  = (float*)d_ws;
  uint32_t* hist  = (uint32_t*)((char*)d_ws + HIST_OFF);
  uint32_t* state = (uint32_t*)((char*)d_ws + STATE_OFF);
  float*    part  = (float*)((char*)d_ws + PART_OFF);

  zero_hist<<<1, 256, 0, stream>>>(hist);

  for (int b = 0; b < BATCH; ++b) {
    const float* ib = inp + (size_t)b * NDIM * DDIM;
    const float* tb = tgt + (size_t)b * NDIM * DDIM;

    // Both sim matrices for this batch (z: 0=input, 1=target)
    gemm_sim<<<dim3(NDIM / 128, NDIM / 128, 2), 256, GEMM_LDS_BYTES, stream>>>(
        ib, tb, sims, state, kptr);

    // Exact k-th-largest threshold via 3-pass radix select (11/11/10 bits)
    for (int pass = 0; pass < 3; ++pass) {
      hist_pass<<<dim3(256, 2), 256, 0, stream>>>(sims, hist, state, pass);
      select_bin<<<2, 64, 0, stream>>>(hist, state, pass);
    }

    mse_pass<<<MSE_BLOCKS, 256, 0, stream>>>(sims, state, part + (size_t)b * MSE_BLOCKS);
  }

  finalize<<<1, 64, 0, stream>>>(part, out);
}